// Post_Process_4243427688550
// MI455X (gfx1250) — compile-verified
//
#include <hip/hip_runtime.h>
#include <hip/hip_bf16.h>

#define NBATCH     16
#define NCLS       80
#define N0         19200
#define N1         4800
#define N2         1200
#define NTOT       25200
#define KMAX       25
#define CAP        4096          // direct-path LDS candidate capacity
#define CAPG       3584          // compact-path global list capacity (mean 3055, +10 sigma)
#define SCORE_T    0.55f
#define IOU_T      0.5f
#define NEGV       (-1e30f)

// -------------------------------------------------------------------------
// Block-wide argmax with lowest-index tie-break (matches jnp.argmax /
// lax.top_k ordering). Result broadcast to all threads; uniform control flow.
// -------------------------------------------------------------------------
__device__ __forceinline__ void block_argmax(float v, int idx,
                                             volatile float* r_val, volatile int* r_idx,
                                             float& out_v, int& out_i) {
    #pragma unroll
    for (int off = 16; off > 0; off >>= 1) {       // wave32 shuffle reduction
        float ov = __shfl_down(v, off);
        int   oi = __shfl_down(idx, off);
        bool take = (oi >= 0) && (idx < 0 || ov > v || (ov == v && oi < idx));
        if (take) { v = ov; idx = oi; }
    }
    const int lane = threadIdx.x & 31;
    const int wid  = threadIdx.x >> 5;
    if (lane == 0) { r_val[wid] = v; r_idx[wid] = idx; }
    __syncthreads();
    if (threadIdx.x == 0) {
        float bv = r_val[0]; int bi = r_idx[0];
        const int nw = (blockDim.x + 31) >> 5;
        for (int w = 1; w < nw; ++w) {
            float ov = r_val[w]; int oi = r_idx[w];
            bool take = (oi >= 0) && (bi < 0 || ov > bv || (ov == bv && oi < bi));
            if (take) { bv = ov; bi = oi; }
        }
        r_val[0] = bv; r_idx[0] = bi;
    }
    __syncthreads();
    out_v = r_val[0];
    out_i = r_idx[0];
    __syncthreads();   // protect scratch before next call
}

// -------------------------------------------------------------------------
// Greedy per-class NMS over an LDS-resident candidate set.
// s_sc: working scores (set NEGV when suppressed); s_b0..3: ORIGINAL corner
// order (post max(,0)); canonicalization happens inside the IoU test, as in
// the reference. Writes 25 keep slots (score=NEGV when invalid).
// -------------------------------------------------------------------------
__device__ __forceinline__ void greedy_nms(float* s_sc, float* s_b0, float* s_b1,
                                           float* s_b2, float* s_b3, int m,
                                           float* __restrict__ ks, float* __restrict__ kb,
                                           volatile float* r_val, volatile int* r_idx,
                                           volatile float* s_sel, volatile int* s_seli) {
    for (int it = 0; it < KMAX; ++it) {
        float lv = NEGV; int li = -1;
        for (int j = threadIdx.x; j < m; j += blockDim.x) {
            float v = s_sc[j];
            if (v > lv) { lv = v; li = j; }        // lowest j per thread on ties
        }
        float bv; int bi;
        block_argmax(lv, li, r_val, r_idx, bv, bi);

        if (threadIdx.x == 0) {
            if (bi >= 0 && bv > SCORE_T) {
                float y0 = s_b0[bi], x0 = s_b1[bi], y1 = s_b2[bi], x1 = s_b3[bi];
                ks[it] = bv;
                kb[it * 4 + 0] = y0; kb[it * 4 + 1] = x0;
                kb[it * 4 + 2] = y1; kb[it * 4 + 3] = x1;
                s_sel[0] = fminf(y0, y1); s_sel[1] = fmaxf(y0, y1);
                s_sel[2] = fminf(x0, x1); s_sel[3] = fmaxf(x0, x1);
                *s_seli = bi;
                s_sc[bi] = NEGV;
            } else {
                *s_seli = -1;
                ks[it] = NEGV;
                kb[it * 4 + 0] = 0.0f; kb[it * 4 + 1] = 0.0f;
                kb[it * 4 + 2] = 0.0f; kb[it * 4 + 3] = 0.0f;
            }
        }
        __syncthreads();

        if (*s_seli >= 0) {
            const float sy1 = s_sel[0], sy2 = s_sel[1], sx1 = s_sel[2], sx2 = s_sel[3];
            const float sarea = (sy2 - sy1) * (sx2 - sx1);
            for (int j = threadIdx.x; j < m; j += blockDim.x) {
                float y0 = s_b0[j], x0 = s_b1[j], y1 = s_b2[j], x1 = s_b3[j];
                float cy1 = fminf(y0, y1), cy2 = fmaxf(y0, y1);
                float cx1 = fminf(x0, x1), cx2 = fmaxf(x0, x1);
                float ih = fmaxf(fminf(sy2, cy2) - fmaxf(sy1, cy1), 0.0f);
                float iw = fmaxf(fminf(sx2, cx2) - fmaxf(sx1, cx1), 0.0f);
                float inter = ih * iw;
                float uni = sarea + (cy2 - cy1) * (cx2 - cx1) - inter;
                float iou = (uni > 0.0f) ? (inter / uni) : 0.0f;
                if (iou > IOU_T) s_sc[j] = NEGV;
            }
        }
        __syncthreads();
    }
}

// =========================================================================
// FAST PATH kernel A: coalesced score scan + scatter to per-(b,c) candidate
// lists. Lane-consecutive elements are class-consecutive -> fully coalesced
// 128B cls lines, and 32 different atomic counters per wave (no contention).
// =========================================================================
__global__ __launch_bounds__(256) void scatter_kernel(
    const float* __restrict__ conf0, const float* __restrict__ cls0,
    const float* __restrict__ conf1, const float* __restrict__ cls1,
    const float* __restrict__ conf2, const float* __restrict__ cls2,
    int* __restrict__ cnt, float* __restrict__ cand_score, int* __restrict__ cand_idx)
{
    const int b = blockIdx.y;
    const int strideE = gridDim.x * blockDim.x;
    const int totalE = NTOT * NCLS;                 // 2,016,000 per batch
    for (int e = blockIdx.x * blockDim.x + threadIdx.x; e < totalE; e += strideE) {
        const int n = e / NCLS;
        const int c = e - n * NCLS;
        const float* cf; const float* cl; int loc;
        if (n < N0) {
            loc = n;
            cf = conf0 + (size_t)b * N0;
            cl = cls0  + (size_t)b * N0 * NCLS;
        } else if (n < N0 + N1) {
            loc = n - N0;
            cf = conf1 + (size_t)b * N1;
            cl = cls1  + (size_t)b * N1 * NCLS;
        } else {
            loc = n - N0 - N1;
            cf = conf2 + (size_t)b * N2;
            cl = cls2  + (size_t)b * N2 * NCLS;
        }
        const float* p = cl + (size_t)loc * NCLS + c;
        __builtin_prefetch(p + strideE, 0, 0);       // gfx1250 global_prefetch_b8
        float s = cf[loc] * (*p);
        if (s > SCORE_T) {
            const int bc = b * NCLS + c;
            int pos = atomicAdd(&cnt[bc], 1);
            if (pos < CAPG) {
                cand_score[(size_t)bc * CAPG + pos] = s;
                cand_idx  [(size_t)bc * CAPG + pos] = n;
            }
        }
    }
}

// =========================================================================
// FAST PATH kernel B: per (b,c) block. Async-load candidate list into LDS
// (ASYNCcnt path), gather boxes by index, then LDS-resident greedy NMS.
// =========================================================================
__global__ __launch_bounds__(256) void nms_compact_kernel(
    const float* __restrict__ bbox0, const float* __restrict__ bbox1,
    const float* __restrict__ bbox2,
    const int* __restrict__ cnt, const float* __restrict__ cand_score,
    const int* __restrict__ cand_idx,
    float* __restrict__ keep_score, float* __restrict__ keep_box)
{
    const int bc = blockIdx.x;
    const int b  = bc / NCLS;

    extern __shared__ float smem[];
    float* s_sc = smem;             // [CAPG]
    float* s_b0 = smem + 1 * CAPG;  // [CAPG] (temporarily holds idx bits)
    float* s_b1 = smem + 2 * CAPG;
    float* s_b2 = smem + 3 * CAPG;
    float* s_b3 = smem + 4 * CAPG;

    __shared__ float r_val[8];
    __shared__ int   r_idx[8];
    __shared__ float s_sel[4];
    __shared__ int   s_seli;

    int m = cnt[bc];
    if (m > CAPG) m = CAPG;

    const float* gsc = cand_score + (size_t)bc * CAPG;
    const int*   gix = cand_idx   + (size_t)bc * CAPG;

    // async global -> LDS: scores into s_sc, indices (raw bits) into s_b0
    for (int i = threadIdx.x; i < m; i += blockDim.x) {
        unsigned           la = (unsigned)(size_t)&s_sc[i];
        unsigned long long ga = (unsigned long long)(gsc + i);
        asm volatile("global_load_async_to_lds_b32 %0, %1, off"
                     :: "v"(la), "v"(ga) : "memory");
        unsigned           lb = (unsigned)(size_t)&s_b0[i];
        unsigned long long gb = (unsigned long long)(gix + i);
        asm volatile("global_load_async_to_lds_b32 %0, %1, off"
                     :: "v"(lb), "v"(gb) : "memory");
    }
    asm volatile("s_wait_asynccnt 0" ::: "memory");
    __syncthreads();

    // gather boxes by candidate index (L2-resident random 16B reads)
    for (int i = threadIdx.x; i < m; i += blockDim.x) {
        int n = ((const int*)s_b0)[i];
        const float* bb; int loc;
        if (n < N0)           { bb = bbox0 + (size_t)b * N0 * 4; loc = n; }
        else if (n < N0 + N1) { bb = bbox1 + (size_t)b * N1 * 4; loc = n - N0; }
        else                  { bb = bbox2 + (size_t)b * N2 * 4; loc = n - N0 - N1; }
        float y0 = fmaxf(bb[loc * 4 + 0], 0.0f);
        float x0 = fmaxf(bb[loc * 4 + 1], 0.0f);
        float y1 = fmaxf(bb[loc * 4 + 2], 0.0f);
        float x1 = fmaxf(bb[loc * 4 + 3], 0.0f);
        s_b0[i] = y0; s_b1[i] = x0; s_b2[i] = y1; s_b3[i] = x1;
    }
    __syncthreads();

    float* ks = keep_score + (size_t)bc * KMAX;
    float* kb = keep_box   + (size_t)bc * KMAX * 4;
    greedy_nms(s_sc, s_b0, s_b1, s_b2, s_b3, m, ks, kb, r_val, r_idx, s_sel, &s_seli);
}

// =========================================================================
// FALLBACK kernel (small workspace): direct per-(b,c) scan + LDS compaction
// + greedy NMS (previous round's design; 640 KB workspace only).
// =========================================================================
__global__ __launch_bounds__(256) void nms_direct_kernel(
    const float* __restrict__ bbox0, const float* __restrict__ conf0, const float* __restrict__ cls0,
    const float* __restrict__ bbox1, const float* __restrict__ conf1, const float* __restrict__ cls1,
    const float* __restrict__ bbox2, const float* __restrict__ conf2, const float* __restrict__ cls2,
    float* __restrict__ keep_score, float* __restrict__ keep_box)
{
    const int b = blockIdx.x / NCLS;
    const int c = blockIdx.x % NCLS;

    extern __shared__ float smem[];
    float* s_sc = smem;
    float* s_b0 = smem + 1 * CAP;
    float* s_b1 = smem + 2 * CAP;
    float* s_b2 = smem + 3 * CAP;
    float* s_b3 = smem + 4 * CAP;

    __shared__ int   s_cnt;
    __shared__ float r_val[8];
    __shared__ int   r_idx[8];
    __shared__ float s_sel[4];
    __shared__ int   s_seli;

    if (threadIdx.x == 0) s_cnt = 0;
    __syncthreads();

    for (int n = threadIdx.x; n < NTOT; n += blockDim.x) {
        const float* bb; const float* cf; const float* cl; int loc;
        if (n < N0) {
            loc = n;
            bb = bbox0 + (size_t)b * N0 * 4;
            cf = conf0 + (size_t)b * N0;
            cl = cls0  + (size_t)b * N0 * NCLS;
        } else if (n < N0 + N1) {
            loc = n - N0;
            bb = bbox1 + (size_t)b * N1 * 4;
            cf = conf1 + (size_t)b * N1;
            cl = cls1  + (size_t)b * N1 * NCLS;
        } else {
            loc = n - N0 - N1;
            bb = bbox2 + (size_t)b * N2 * 4;
            cf = conf2 + (size_t)b * N2;
            cl = cls2  + (size_t)b * N2 * NCLS;
        }
        __builtin_prefetch(cl + (size_t)(loc + blockDim.x) * NCLS + c, 0, 0);
        float score = cf[loc] * cl[(size_t)loc * NCLS + c];
        if (score > SCORE_T) {
            float y0 = fmaxf(bb[loc * 4 + 0], 0.0f);
            float x0 = fmaxf(bb[loc * 4 + 1], 0.0f);
            float y1 = fmaxf(bb[loc * 4 + 2], 0.0f);
            float x1 = fmaxf(bb[loc * 4 + 3], 0.0f);
            int k = atomicAdd(&s_cnt, 1);
            if (k < CAP) {
                s_sc[k] = score;
                s_b0[k] = y0; s_b1[k] = x0; s_b2[k] = y1; s_b3[k] = x1;
            }
        }
    }
    __syncthreads();
    const int m = (s_cnt < CAP) ? s_cnt : CAP;

    float* ks = keep_score + ((size_t)b * NCLS + c) * KMAX;
    float* kb = keep_box   + ((size_t)b * NCLS + c) * KMAX * 4;
    greedy_nms(s_sc, s_b0, s_b1, s_b2, s_b3, m, ks, kb, r_val, r_idx, s_sel, &s_seli);
}

// =========================================================================
// Kernel C: per-batch top-25 over the 80x25 keep table (async LDS gather).
// Output (floats): boxes[16][25][4] | scores[16][25] | cls[16][25] | valid[16]
// =========================================================================
__global__ __launch_bounds__(256) void topk_kernel(
    const float* __restrict__ keep_score, const float* __restrict__ keep_box,
    float* __restrict__ out)
{
    const int b = blockIdx.x;
    const int M = NCLS * KMAX;   // 2000

    __shared__ float               s_sc[NCLS * KMAX];
    __shared__ __align__(16) float s_bx[NCLS * KMAX * 4];
    __shared__ float r_val[8];
    __shared__ int   r_idx[8];

    const float* gs = keep_score + (size_t)b * M;
    const float* gb = keep_box   + (size_t)b * M * 4;

    for (int i = threadIdx.x; i < M; i += blockDim.x) {
        unsigned           la = (unsigned)(size_t)&s_sc[i];
        unsigned long long ga = (unsigned long long)(gs + i);
        asm volatile("global_load_async_to_lds_b32 %0, %1, off"
                     :: "v"(la), "v"(ga) : "memory");
        unsigned           lb = (unsigned)(size_t)&s_bx[i * 4];
        unsigned long long gb4 = (unsigned long long)(gb + (size_t)i * 4);
        asm volatile("global_load_async_to_lds_b128 %0, %1, off"
                     :: "v"(lb), "v"(gb4) : "memory");
    }
    asm volatile("s_wait_asynccnt 0" ::: "memory");
    __syncthreads();

    int vc = 0;
    for (int k = 0; k < KMAX; ++k) {
        float lv = NEGV; int li = -1;
        for (int j = threadIdx.x; j < M; j += blockDim.x) {
            float v = s_sc[j];
            if (v > lv) { lv = v; li = j; }
        }
        float bv; int bi;
        block_argmax(lv, li, r_val, r_idx, bv, bi);

        if (threadIdx.x == 0) {
            s_sc[bi] = NEGV;
            const bool valid = bv > SCORE_T;
            float o0 = 0.f, o1 = 0.f, o2 = 0.f, o3 = 0.f, osc = 0.f, ocl = 0.f;
            if (valid) {
                o0 = fminf(fmaxf(s_bx[bi * 4 + 0], 0.0f), 1.0f);
                o1 = fminf(fmaxf(s_bx[bi * 4 + 1], 0.0f), 1.0f);
                o2 = fminf(fmaxf(s_bx[bi * 4 + 2], 0.0f), 1.0f);
                o3 = fminf(fmaxf(s_bx[bi * 4 + 3], 0.0f), 1.0f);
                osc = bv;
                ocl = (float)(bi / KMAX);
                vc++;
            }
            const int slot = b * KMAX + k;
            out[slot * 4 + 0] = o0;
            out[slot * 4 + 1] = o1;
            out[slot * 4 + 2] = o2;
            out[slot * 4 + 3] = o3;
            out[NBATCH * KMAX * 4 + slot] = osc;
            out[NBATCH * KMAX * 4 + NBATCH * KMAX + slot] = ocl;
        }
        __syncthreads();
    }
    if (threadIdx.x == 0) {
        out[NBATCH * KMAX * 4 + 2 * NBATCH * KMAX + b] = (float)vc;
    }
}

// -------------------------------------------------------------------------
extern "C" void kernel_launch(void* const* d_in, const int* in_sizes, int n_in,
                              void* d_out, int out_size, void* d_ws, size_t ws_size,
                              hipStream_t stream) {
    const float* bbox0 = (const float*)d_in[0];
    const float* conf0 = (const float*)d_in[1];
    const float* cls0  = (const float*)d_in[2];
    const float* bbox1 = (const float*)d_in[3];
    const float* conf1 = (const float*)d_in[4];
    const float* cls1  = (const float*)d_in[5];
    const float* bbox2 = (const float*)d_in[6];
    const float* conf2 = (const float*)d_in[7];
    const float* cls2  = (const float*)d_in[8];

    // --- workspace layout (shared by both paths) ---
    float* keep_score = (float*)d_ws;                                  // 128 KB
    float* keep_box   = keep_score + (size_t)NBATCH * NCLS * KMAX;     // 512 KB
    // fast-path extras
    char*  base       = (char*)d_ws;
    int*   cnt        = (int*)(base + 656 * 1024);                     // 5 KB (padded)
    float* cand_score = (float*)(base + 664 * 1024);
    int*   cand_idx   = (int*)((char*)cand_score + (size_t)NBATCH * NCLS * CAPG * sizeof(float));
    const size_t need_fast = 664 * 1024 + (size_t)NBATCH * NCLS * CAPG * 8;

    if (ws_size >= need_fast) {
        // FAST PATH: coalesced scatter + async-LDS compact NMS (~6x less L2 traffic)
        hipMemsetAsync(cnt, 0, NBATCH * NCLS * sizeof(int), stream);
        scatter_kernel<<<dim3(2048, NBATCH), 256, 0, stream>>>(
            conf0, cls0, conf1, cls1, conf2, cls2, cnt, cand_score, cand_idx);
        nms_compact_kernel<<<NBATCH * NCLS, 256, (size_t)CAPG * 5 * sizeof(float), stream>>>(
            bbox0, bbox1, bbox2, cnt, cand_score, cand_idx, keep_score, keep_box);
    } else {
        // FALLBACK: direct strided scan (only 640 KB workspace required)
        nms_direct_kernel<<<NBATCH * NCLS, 256, (size_t)CAP * 5 * sizeof(float), stream>>>(
            bbox0, conf0, cls0, bbox1, conf1, cls1, bbox2, conf2, cls2,
            keep_score, keep_box);
    }

    topk_kernel<<<NBATCH, 256, 0, stream>>>(keep_score, keep_box, (float*)d_out);
}